// AutoEncoder_13108240187605
// MI455X (gfx1250) — compile-verified
//
#include <hip/hip_runtime.h>
#include <hip/hip_bf16.h>
#include <math.h>

typedef float v2f __attribute__((ext_vector_type(2)));
typedef float v8f __attribute__((ext_vector_type(8)));

#define B_    128
#define HW    65536     // 256*256
#define NENC  82        // EMB+18
#define NPAD  96        // padded to 6x16
#define EMB_  64
#define KSLICES 128
#define KS    512       // 65536 / 128

// ---------------------------------------------------------------------------
// Encoder split-K partial GEMM: partial[s] = x[16 rows, K-slice] @ W_enc[K-slice]
// One wave per block. grid = (8 m-tiles, 128 k-slices).
// Branch-free inner loop: padded columns (82..95) use a clamped address and a
// per-lane 0/1 mask so EXEC stays all-ones (WMMA requirement) and loads clause.
// ---------------------------------------------------------------------------
__global__ __launch_bounds__(32)
void enc_partial_kernel(const float* __restrict__ x,
                        const float* __restrict__ We,
                        float* __restrict__ partial) {
    const int lane = threadIdx.x & 31;
    const int lm   = lane & 15;
    const int half = lane >> 4;
    const int m0   = blockIdx.x * 16;
    const int k0   = blockIdx.y * KS;

    v8f acc[6];
#pragma unroll
    for (int nt = 0; nt < 6; ++nt) acc[nt] = (v8f){0,0,0,0,0,0,0,0};

    // Per-lane column index + validity mask per N-tile (hoisted; tiles 0..4
    // are always valid, tile 5 is partially masked).
    int   ncol[6];
    float nmask[6];
#pragma unroll
    for (int nt = 0; nt < 6; ++nt) {
        const int n = nt * 16 + lm;
        ncol[nt]  = (n < NENC) ? n : (NENC - 1);   // clamped: always in-bounds
        nmask[nt] = (n < NENC) ? 1.0f : 0.0f;
    }

    const float* xrow = x + (m0 + lm) * HW;

    for (int k = k0; k < k0 + KS; k += 4) {
        const int kb = k + half * 2;
        v2f a;
        a.x = xrow[kb];
        a.y = xrow[kb + 1];
        const float* Wr0 = We + (size_t)kb * NENC;
#pragma unroll
        for (int nt = 0; nt < 6; ++nt) {
            v2f bf;
            bf.x = Wr0[ncol[nt]]        * nmask[nt];
            bf.y = Wr0[NENC + ncol[nt]] * nmask[nt];
            acc[nt] = __builtin_amdgcn_wmma_f32_16x16x4_f32(
                false, a, false, bf, (short)0, acc[nt], false, false);
        }
    }

    float* pout = partial + (size_t)blockIdx.y * (B_ * NPAD);
#pragma unroll
    for (int nt = 0; nt < 6; ++nt) {
#pragma unroll
        for (int r = 0; r < 8; ++r) {
            const int row = m0 + r + half * 8;
            const int col = nt * 16 + lm;
            pout[row * NPAD + col] = acc[nt][r];
        }
    }
}

// ---------------------------------------------------------------------------
// Fixed-order split-K reduction + bias -> feats[128 x 96]  (deterministic)
// ---------------------------------------------------------------------------
__global__ __launch_bounds__(256)
void reduce_feats_kernel(const float* __restrict__ partial,
                         const float* __restrict__ be,
                         float* __restrict__ feats) {
    const int i = blockIdx.x * blockDim.x + threadIdx.x;   // 0 .. 128*96-1
    if (i >= B_ * NPAD) return;
    const int j = i % NPAD;
    float s = (j < NENC) ? be[j] : 0.0f;
#pragma unroll 4
    for (int sl = 0; sl < KSLICES; ++sl)
        s += partial[(size_t)sl * (B_ * NPAD) + i];
    feats[i] = s;
}

// ---------------------------------------------------------------------------
// Decoder GEMM: out = emb(128x64) @ W_dec(64x65536) + b_dec
// One wave per block, 16x64 output tile. grid = (8 m-tiles, 1024 pixel blocks)
// ---------------------------------------------------------------------------
__global__ __launch_bounds__(32)
void dec_kernel(const float* __restrict__ feats,
                const float* __restrict__ Wd,
                const float* __restrict__ bd,
                float* __restrict__ out) {
    const int lane = threadIdx.x & 31;
    const int lm   = lane & 15;
    const int half = lane >> 4;
    const int m0   = blockIdx.x * 16;
    const int p0   = blockIdx.y * 64;

    v8f acc[4];
#pragma unroll
    for (int nt = 0; nt < 4; ++nt) acc[nt] = (v8f){0,0,0,0,0,0,0,0};

    const float* frow = feats + (m0 + lm) * NPAD;

#pragma unroll
    for (int k = 0; k < EMB_; k += 4) {
        const int kb = k + half * 2;
        v2f a;
        a.x = frow[kb];
        a.y = frow[kb + 1];
#pragma unroll
        for (int nt = 0; nt < 4; ++nt) {
            const int p = p0 + nt * 16 + lm;
            v2f bf;
            bf.x = Wd[(size_t)kb * HW + p];
            bf.y = Wd[(size_t)(kb + 1) * HW + p];
            acc[nt] = __builtin_amdgcn_wmma_f32_16x16x4_f32(
                false, a, false, bf, (short)0, acc[nt], false, false);
        }
    }

#pragma unroll
    for (int nt = 0; nt < 4; ++nt) {
#pragma unroll
        for (int r = 0; r < 8; ++r) {
            const int row = m0 + r + half * 8;
            const int p   = p0 + nt * 16 + lm;
            out[(size_t)row * HW + p] = acc[nt][r] + bd[p];
        }
    }
}

// ---------------------------------------------------------------------------
// Affine-grid bilinear sampling pass (zero padding outside image).
// theta = feats[b, 64+toff .. 64+toff+5].  grid = (256 pixel chunks, 128 batch)
// ---------------------------------------------------------------------------
__global__ __launch_bounds__(256)
void sample_kernel(const float* __restrict__ src,
                   const float* __restrict__ feats,
                   float* __restrict__ dst,
                   int toff) {
    const int b = blockIdx.y;
    const int p = blockIdx.x * blockDim.x + threadIdx.x;   // 0..65535
    const int h = p >> 8;
    const int w = p & 255;

    const float* th = feats + b * NPAD + EMB_ + toff;
    const float t00 = th[0], t01 = th[1], t02 = th[2];
    const float t10 = th[3], t11 = th[4], t12 = th[5];

    const float X = (float)(w * 2 + 1) * (1.0f / 256.0f) - 1.0f;
    const float Y = (float)(h * 2 + 1) * (1.0f / 256.0f) - 1.0f;
    const float gx = t00 * X + t01 * Y + t02;
    const float gy = t10 * X + t11 * Y + t12;

    const float ix = (gx + 1.0f) * 128.0f - 0.5f;
    const float iy = (gy + 1.0f) * 128.0f - 0.5f;
    const float ix0f = floorf(ix);
    const float iy0f = floorf(iy);
    const int ix0 = (int)ix0f;
    const int iy0 = (int)iy0f;
    const float wx1 = ix - ix0f, wx0 = 1.0f - wx1;
    const float wy1 = iy - iy0f, wy0 = 1.0f - wy1;

    const float* img = src + (size_t)b * HW;

    auto gather = [&](int xx, int yy) -> float {
        const bool valid = (xx >= 0) & (xx < 256) & (yy >= 0) & (yy < 256);
        const int xc = min(max(xx, 0), 255);
        const int yc = min(max(yy, 0), 255);
        const float v = img[yc * 256 + xc];
        return valid ? v : 0.0f;
    };

    const float v = gather(ix0,     iy0    ) * (wx0 * wy0)
                  + gather(ix0 + 1, iy0    ) * (wx1 * wy0)
                  + gather(ix0,     iy0 + 1) * (wx0 * wy1)
                  + gather(ix0 + 1, iy0 + 1) * (wx1 * wy1);

    dst[(size_t)b * HW + p] = v;
}

// ---------------------------------------------------------------------------
extern "C" void kernel_launch(void* const* d_in, const int* in_sizes, int n_in,
                              void* d_out, int out_size, void* d_ws, size_t ws_size,
                              hipStream_t stream) {
    const float* x    = (const float*)d_in[0];
    const float* We   = (const float*)d_in[1];
    const float* be   = (const float*)d_in[2];
    const float* Wd   = (const float*)d_in[3];
    const float* bd   = (const float*)d_in[4];
    float*       out  = (float*)d_out;

    // Workspace layout:
    //   [0, 33.5MB)      : image ping buffer (ws_img)
    //   [+33.5MB, +6.3MB): encoder split-K partials (128 x 128 x 96 f32)
    //   [.., +48KB)      : feats (128 x 96 f32)
    char* base = (char*)d_ws;
    float* ws_img  = (float*)base;
    float* partial = (float*)(base + (size_t)B_ * HW * sizeof(float));
    float* feats   = partial + (size_t)KSLICES * B_ * NPAD;

    // 1) encoder split-K partial GEMM (WMMA f32 16x16x4)
    enc_partial_kernel<<<dim3(B_ / 16, KSLICES), 32, 0, stream>>>(x, We, partial);

    // 2) deterministic reduction + bias
    reduce_feats_kernel<<<(B_ * NPAD + 255) / 256, 256, 0, stream>>>(partial, be, feats);

    // 3) decoder GEMM (WMMA f32 16x16x4) -> ws_img
    dec_kernel<<<dim3(B_ / 16, HW / 64), 32, 0, stream>>>(feats, Wd, bd, ws_img);

    // 4) grid_sample with translation (feats[:,76:82]) : ws_img -> d_out
    sample_kernel<<<dim3(HW / 256, B_), 256, 0, stream>>>(ws_img, feats, out, 12);
    // 5) grid_sample with scaler_shear (feats[:,64:70]) : d_out -> ws_img
    sample_kernel<<<dim3(HW / 256, B_), 256, 0, stream>>>(out, feats, ws_img, 0);
    // 6) grid_sample with rotation (feats[:,70:76])     : ws_img -> d_out
    sample_kernel<<<dim3(HW / 256, B_), 256, 0, stream>>>(ws_img, feats, out, 6);
}